// LogicMessagePassingNetwork_11003706213189
// MI455X (gfx1250) — compile-verified
//
#include <hip/hip_runtime.h>

#define D 64
#define NREL 50
#define TP 66   // padded LDS row stride (floats) for the A tile

typedef float v2f __attribute__((ext_vector_type(2)));
typedef float v8f __attribute__((ext_vector_type(8)));

// ---------------------------------------------------------------------------
// Kernel 1: fold rel_emb through the weight halves once (50 distinct h rows).
//   tabA = rel_emb @ W_msg[0:64,:]   tabB = rel_emb @ W_msg[64:128,:]
//   tabC = rel_emb @ W_upd[0:64,:]
// ---------------------------------------------------------------------------
__global__ __launch_bounds__(64) void build_tables_kernel(
    const float* __restrict__ rel_emb,
    const float* __restrict__ W_msg,
    const float* __restrict__ W_upd,
    float* __restrict__ tabA, float* __restrict__ tabB, float* __restrict__ tabC)
{
    const int r = blockIdx.x;   // relation id
    const int j = threadIdx.x;  // output column
    float a = 0.f, b = 0.f, c = 0.f;
#pragma unroll 8
    for (int k = 0; k < D; ++k) {
        const float e = rel_emb[r * D + k];
        a += e * W_msg[k * D + j];
        b += e * W_msg[(D + k) * D + j];
        c += e * W_upd[k * D + j];
    }
    tabA[r * D + j] = a;
    tabB[r * D + j] = b;
    tabC[r * D + j] = c;
}

// ---------------------------------------------------------------------------
// Kernel 2: triangle messages + scatter-add.
// One wave per triangle iteration; lane handles 2 of the 64 dims.
//   msg = relu(tabA[rel[ab]] + tabB[rel[bc]]);  agg[ac] += msg (atomic f32)
// ---------------------------------------------------------------------------
__global__ __launch_bounds__(256) void tri_scatter_kernel(
    const int* __restrict__ eab, const int* __restrict__ ebc,
    const int* __restrict__ eac, const int* __restrict__ rel,
    const float* __restrict__ tabA, const float* __restrict__ tabB,
    float* __restrict__ agg, int T)
{
    const int lane = threadIdx.x & 31;
    const int wid  = (blockIdx.x << 3) + (threadIdx.x >> 5);
    const int nw   = gridDim.x << 3;
    for (int t = wid; t < T; t += nw) {
        const int ra = rel[eab[t]];
        const int rb = rel[ebc[t]];
        const int ea = eac[t];
        const float2 va = ((const float2*)(tabA + ra * D))[lane];
        const float2 vb = ((const float2*)(tabB + rb * D))[lane];
        float mx = va.x + vb.x; mx = mx > 0.f ? mx : 0.f;
        float my = va.y + vb.y; my = my > 0.f ? my : 0.f;
        float* p = agg + (size_t)ea * D + lane * 2;
        unsafeAtomicAdd(p,     mx);   // global_atomic_add_f32 (no CAS loop)
        unsafeAtomicAdd(p + 1, my);
    }
}

// ---------------------------------------------------------------------------
// Kernel 3: out = relu(tabC[rel] + agg @ W_upd[64:128,:]) in place over d_out.
// 8 waves/block, 16 rows per wave, full N=64 per wave.
// Fragment layouts for V_WMMA_F32_16X16X4_F32 (wave32):
//   A: lane L -> M=L%16, K = k0 + 2*(L/16) + j   (j = vector elem)
//   B: lane L -> N=L%16, K = k0 + 2*(L/16) + j
//   D: elem r -> M = r + 8*(L/16), N = L%16
// Wub staged in LDS with paired-row interleave sWp[(k>>1)*128 + n*2 + (k&1)]
// so each B fragment is one contiguous 8-byte LDS read (single ds_load_b64).
// ---------------------------------------------------------------------------
__global__ __launch_bounds__(256) void upd_gemm_kernel(
    float* __restrict__ out,            // in: agg, out: result  [E, 64]
    const float* __restrict__ W_upd,    // [128, 64]; bottom 64 rows used
    const float* __restrict__ tabC,     // [NREL, 64]
    const int* __restrict__ rel,        // [E]
    int E)
{
    __shared__ float sWp[D * D];        // 16 KB: Wub, paired-row interleaved
    __shared__ float sA[8][16 * TP];    // ~33 KB: per-wave 16x64 agg tile

    const int tid  = threadIdx.x;
    const int wave = tid >> 5;
    const int lane = tid & 31;
    const int half = lane >> 4;         // 0 | 1
    const int l16  = lane & 15;

    // Stage W_upd rows 64..127 into paired-row interleaved layout:
    //   sWp[(k>>1)*128 + n*2 + (k&1)] = Wub[k][n]
    {
        const int k  = tid >> 2;            // 0..63
        const int nb = (tid & 3) << 4;      // 0,16,32,48
        const float* src = W_upd + D * D + k * D + nb;
        float* dst = sWp + (k >> 1) * 128 + (k & 1) + nb * 2;
#pragma unroll
        for (int i = 0; i < 16; ++i)
            dst[i * 2] = src[i];
    }

    const long long m0 = (long long)blockIdx.x * 128 + wave * 16;

    // Stage this wave's 16x64 tile of agg (one coalesced row per step).
#pragma unroll
    for (int r = 0; r < 16; ++r) {
        long long row = m0 + r;
        if (row >= E) row = E - 1;      // clamp: loads harmless, stores guarded
        const float2 v = ((const float2*)(out + row * D))[lane];
        *(float2*)&sA[wave][r * TP + lane * 2] = v;
    }

    __syncthreads();

    v8f acc0 = {}, acc1 = {}, acc2 = {}, acc3 = {};

#pragma unroll
    for (int k0 = 0; k0 < D; k0 += 4) {
        const int ka = k0 + 2 * half;                 // even -> 8B-aligned reads
        const int kp = (k0 >> 1) + half;              // paired-row index
        const v2f a  = *(const v2f*)&sA[wave][l16 * TP + ka];
        const float* wb = sWp + kp * 128 + l16 * 2;
        const v2f b0 = *(const v2f*)(wb +  0);        // N-tile 0 (cols  0..15)
        const v2f b1 = *(const v2f*)(wb + 32);        // N-tile 1 (cols 16..31)
        const v2f b2 = *(const v2f*)(wb + 64);        // N-tile 2 (cols 32..47)
        const v2f b3 = *(const v2f*)(wb + 96);        // N-tile 3 (cols 48..63)
        acc0 = __builtin_amdgcn_wmma_f32_16x16x4_f32(false, a, false, b0, (short)0, acc0, false, false);
        acc1 = __builtin_amdgcn_wmma_f32_16x16x4_f32(false, a, false, b1, (short)0, acc1, false, false);
        acc2 = __builtin_amdgcn_wmma_f32_16x16x4_f32(false, a, false, b2, (short)0, acc2, false, false);
        acc3 = __builtin_amdgcn_wmma_f32_16x16x4_f32(false, a, false, b3, (short)0, acc3, false, false);
    }

    // Epilogue: add tabC[rel[m]], ReLU, store (per-lane guard; WMMA done).
#pragma unroll
    for (int r = 0; r < 8; ++r) {
        const long long m = m0 + r + 8 * half;
        if (m < E) {
            const int rl = rel[m];
            const float* tc = tabC + rl * D;
            float o0 = acc0[r] + tc[ 0 + l16];
            float o1 = acc1[r] + tc[16 + l16];
            float o2 = acc2[r] + tc[32 + l16];
            float o3 = acc3[r] + tc[48 + l16];
            float* op = out + m * D;
            op[ 0 + l16] = o0 > 0.f ? o0 : 0.f;
            op[16 + l16] = o1 > 0.f ? o1 : 0.f;
            op[32 + l16] = o2 > 0.f ? o2 : 0.f;
            op[48 + l16] = o3 > 0.f ? o3 : 0.f;
        }
    }
}

// ---------------------------------------------------------------------------
extern "C" void kernel_launch(void* const* d_in, const int* in_sizes, int n_in,
                              void* d_out, int out_size, void* d_ws, size_t ws_size,
                              hipStream_t stream)
{
    const float* rel_emb = (const float*)d_in[0];   // [50, 64]
    const float* W_msg   = (const float*)d_in[1];   // [128, 64]
    const float* W_upd   = (const float*)d_in[2];   // [128, 64]
    // d_in[3] = src, d_in[4] = dst : unused by the reference computation
    const int* rel = (const int*)d_in[5];           // [E]
    const int* eab = (const int*)d_in[6];           // [T]
    const int* ebc = (const int*)d_in[7];           // [T]
    const int* eac = (const int*)d_in[8];           // [T]
    float* out = (float*)d_out;                     // [E, 64], doubles as agg

    const int E = in_sizes[5];
    const int T = in_sizes[6];

    float* tabA = (float*)d_ws;                     // 3 * 50*64 f32 = 38.4 KB
    float* tabB = tabA + NREL * D;
    float* tabC = tabB + NREL * D;

    build_tables_kernel<<<NREL, D, 0, stream>>>(rel_emb, W_msg, W_upd, tabA, tabB, tabC);
    hipMemsetAsync(out, 0, (size_t)E * D * sizeof(float), stream);
    tri_scatter_kernel<<<4096, 256, 0, stream>>>(eab, ebc, eac, rel, tabA, tabB, out, T);
    const int nblk = (E + 127) / 128;
    upd_gemm_kernel<<<nblk, 256, 0, stream>>>(out, W_upd, tabC, rel, E);
}